// ShiftGCN_ST_new_50165218018167
// MI455X (gfx1250) — compile-verified
//
#include <hip/hip_runtime.h>
#include <hip/hip_bf16.h>

typedef __attribute__((ext_vector_type(2))) float v2f;
typedef __attribute__((ext_vector_type(8))) float v8f;

#define NBATCH 64
#define CIN    64
#define COUT   64
#define TFR    128
#define VJ     55
#define EPSF   1e-5f

#define NT_TOT (NBATCH * TFR)        // 8192
#define VC     (VJ * COUT)           // 3520
#define Z_ELTS ((long)NT_TOT * VC)   // 28,835,840 floats

// ws layout (floats): [ z : Z_ELTS | scale1 : 3520 | shift1 : 3520 | tmT : 4096 | scale2 : 64 | shift2 : 64 ]

// ---------------------------------------------------------------------------
// Kernel 0: one block. Precompute folded BN affine terms + masked-tanh table.
//   scale1[v*64+d] = g1*rsqrt(var1+eps)
//   shift1[v*64+d] = be1 + scale1*(b[d] - mu1)         (bias folded in)
//   tmT[c*64+row]  = tanh(fm[row*64+c]) + 1            (transposed, coalesced)
//   scale2[d], shift2[d] likewise for BN2.
// ---------------------------------------------------------------------------
__global__ __launch_bounds__(256)
void shiftgcn_precompute(const float* __restrict__ b,
                         const float* __restrict__ fm,
                         const float* __restrict__ g1,  const float* __restrict__ be1,
                         const float* __restrict__ mu1, const float* __restrict__ vr1,
                         const float* __restrict__ g2,  const float* __restrict__ be2,
                         const float* __restrict__ mu2, const float* __restrict__ vr2,
                         float* __restrict__ scale1, float* __restrict__ shift1,
                         float* __restrict__ tmT,
                         float* __restrict__ scale2, float* __restrict__ shift2)
{
    const int tid = threadIdx.x;
    for (int i = tid; i < VC; i += 256) {
        const float s = g1[i] * rsqrtf(vr1[i] + EPSF);
        scale1[i] = s;
        shift1[i] = be1[i] + s * (b[i & 63] - mu1[i]);
    }
    for (int i = tid; i < 64 * 64; i += 256) {
        const int c = i >> 6, row = i & 63;
        tmT[i] = (row < VJ) ? (tanhf(fm[row * CIN + c]) + 1.0f) : 0.0f;
    }
    for (int i = tid; i < COUT; i += 256) {
        const float s = g2[i] * rsqrtf(vr2[i] + EPSF);
        scale2[i] = s;
        shift2[i] = be2[i] - s * mu2[i];
    }
}

// ---------------------------------------------------------------------------
// Kernel 1: per-(n,t) block, 128 threads (4 waves).
//   Stage A[v,c] = x[n,c,t,(v+c)%55] * tmT[c,v] into LDS (coalesced: lanes
//   walk v for fixed c), stage W, then 4 waves x 4 M-tiles x 16 K-steps of
//   V_WMMA_F32_16X16X4_F32. Epilogue applies folded BN1 affine, stores z.
// ---------------------------------------------------------------------------
__global__ __launch_bounds__(128)
void shiftgcn_gemm_bn1(const float* __restrict__ x,
                       const float* __restrict__ W,
                       const float* __restrict__ tmT,
                       const float* __restrict__ scale1,
                       const float* __restrict__ shift1,
                       float* __restrict__ z)
{
    __shared__ float ldsA[64 * 65];
    __shared__ float ldsW[64 * 65];

    const int nt  = blockIdx.x;         // 0..8191
    const int n   = nt >> 7;
    const int t   = nt & 127;
    const int tid = threadIdx.x;

    const long xbase = (long)n * (CIN * TFR * VJ) + (long)t * VJ;

    // ---- stage A: consecutive lanes -> consecutive v (coalesced x reads) ----
    for (int i = tid; i < 64 * 64; i += 128) {
        const int c   = i >> 6;         // channel
        const int row = i & 63;         // v (padded to 64)
        float val = 0.0f;
        if (row < VJ) {
            const int vv = (row + c) % VJ;              // spatial shift gather
            val = x[xbase + (long)c * (TFR * VJ) + vv] * tmT[c * 64 + row];
        }
        ldsA[row * 65 + c] = val;       // bank (row+c)%64: conflict-free
    }
    // ---- stage W (vectorized) ----
    {
        const float4* W4 = (const float4*)W;
        for (int j = tid; j < 1024; j += 128) {
            const float4 w = W4[j];
            const int k = (j * 4) >> 6, d = (j * 4) & 63;
            ldsW[k * 65 + d + 0] = w.x;
            ldsW[k * 65 + d + 1] = w.y;
            ldsW[k * 65 + d + 2] = w.z;
            ldsW[k * 65 + d + 3] = w.w;
        }
    }
    __syncthreads();

    const int lane = tid & 31;
    const int wv   = tid >> 5;          // wave id -> N-tile
    const int half = lane >> 4;         // 0: lanes 0-15, 1: lanes 16-31
    const int l16  = lane & 15;
    const int dcol = wv * 16 + l16;

    v8f acc[4];
    acc[0] = v8f{}; acc[1] = v8f{}; acc[2] = v8f{}; acc[3] = v8f{};

    for (int ks = 0; ks < 16; ++ks) {
        const int kb = ks * 4 + half * 2;   // this lane-half's K pair

        v2f bf;                              // B fragment: W[kb+j][dcol]
        bf.x = ldsW[(kb + 0) * 65 + dcol];
        bf.y = ldsW[(kb + 1) * 65 + dcol];

        v2f a0, a1, a2, a3;                  // A fragments: A[m][kb+j]
        a0.x = ldsA[(l16 +  0) * 65 + kb];  a0.y = ldsA[(l16 +  0) * 65 + kb + 1];
        a1.x = ldsA[(l16 + 16) * 65 + kb];  a1.y = ldsA[(l16 + 16) * 65 + kb + 1];
        a2.x = ldsA[(l16 + 32) * 65 + kb];  a2.y = ldsA[(l16 + 32) * 65 + kb + 1];
        a3.x = ldsA[(l16 + 48) * 65 + kb];  a3.y = ldsA[(l16 + 48) * 65 + kb + 1];

        acc[0] = __builtin_amdgcn_wmma_f32_16x16x4_f32(false, a0, false, bf, (short)0, acc[0], false, false);
        acc[1] = __builtin_amdgcn_wmma_f32_16x16x4_f32(false, a1, false, bf, (short)0, acc[1], false, false);
        acc[2] = __builtin_amdgcn_wmma_f32_16x16x4_f32(false, a2, false, bf, (short)0, acc[2], false, false);
        acc[3] = __builtin_amdgcn_wmma_f32_16x16x4_f32(false, a3, false, bf, (short)0, acc[3], false, false);
    }

    // ---- epilogue: folded BN1 affine, store z ----
    #pragma unroll
    for (int mt = 0; mt < 4; ++mt) {
        #pragma unroll
        for (int i = 0; i < 8; ++i) {
            const int m = half * 8 + i;      // C/D layout: row = half*8 + vgpr
            const int v = mt * 16 + m;
            if (v < VJ) {
                const int idx1 = v * COUT + dcol;
                z[((long)nt * VJ + v) * COUT + dcol] =
                    scale1[idx1] * acc[mt][i] + shift1[idx1];
            }
        }
    }
}

// ---------------------------------------------------------------------------
// Kernel 2: per-(n,t) block, 256 threads.
//   Stage relu(z[n, t-1..t+1, :, :]) into LDS (coalesced float4 reads,
//   zero rows for out-of-range frames), then emit
//   out[n,d,t,v] = relu( scale2[d]*lds[off(d)+1][(v-d)%55, d] + shift2[d]
//                        + x[n,d,t,v] )
//   LDS row stride 65 -> read bank (vs+d)%64: conflict-free over v.
// ---------------------------------------------------------------------------
__global__ __launch_bounds__(256)
void shiftgcn_tail(const float* __restrict__ z,
                   const float* __restrict__ x,
                   const float* __restrict__ scale2,
                   const float* __restrict__ shift2,
                   float* __restrict__ out)
{
    __shared__ float ldsz[3][VJ * 65];   // 3 frames x 55 rows x 65 (padded)

    const int nt  = blockIdx.x;
    const int n   = nt >> 7;
    const int t   = nt & 127;
    const int tid = threadIdx.x;

    // ---- stage 3 temporal frames of z, ReLU'd, zero-padded ----
    #pragma unroll
    for (int dt = 0; dt < 3; ++dt) {
        const int tsrc = t + dt - 1;
        if (tsrc >= 0 && tsrc < TFR) {
            const float4* zr = (const float4*)(z + (long)(n * TFR + tsrc) * VC);
            for (int j = tid; j < VC / 4; j += 256) {
                const float4 p = zr[j];
                const int jj = j * 4;
                const int row = jj >> 6, col = jj & 63;
                float* dst = &ldsz[dt][row * 65 + col];
                dst[0] = fmaxf(p.x, 0.0f);
                dst[1] = fmaxf(p.y, 0.0f);
                dst[2] = fmaxf(p.z, 0.0f);
                dst[3] = fmaxf(p.w, 0.0f);
            }
        } else {
            for (int j = tid; j < VC; j += 256) {
                ldsz[dt][(j >> 6) * 65 + (j & 63)] = 0.0f;
            }
        }
    }
    __syncthreads();

    // ---- compute: i -> (d = i/55, v = i%55); writes contiguous in v ----
    const long obase = (long)n * (COUT * TFR * VJ) + (long)t * VJ;
    for (int i = tid; i < VC; i += 256) {
        const int d = i / VJ;
        const int v = i - d * VJ;
        const int dt = d % 3;            // off = dt-1 -> frame index dt
        int vs = v - d;                  // inverse spatial shift
        vs %= VJ; if (vs < 0) vs += VJ;

        const float gval = ldsz[dt][vs * 65 + d];
        const float r    = scale2[d] * gval + shift2[d];
        const long  xi   = obase + (long)d * (TFR * VJ) + v;
        out[xi] = fmaxf(r + x[xi], 0.0f);
    }
}

// ---------------------------------------------------------------------------
extern "C" void kernel_launch(void* const* d_in, const int* in_sizes, int n_in,
                              void* d_out, int out_size, void* d_ws, size_t ws_size,
                              hipStream_t stream)
{
    const float* x   = (const float*)d_in[0];
    const float* W   = (const float*)d_in[1];
    const float* b   = (const float*)d_in[2];
    const float* fm  = (const float*)d_in[3];
    const float* g1  = (const float*)d_in[4];
    const float* be1 = (const float*)d_in[5];
    const float* mu1 = (const float*)d_in[6];
    const float* vr1 = (const float*)d_in[7];
    const float* g2  = (const float*)d_in[8];
    const float* be2 = (const float*)d_in[9];
    const float* mu2 = (const float*)d_in[10];
    const float* vr2 = (const float*)d_in[11];

    float* ws     = (float*)d_ws;
    float* z      = ws;                         // Z_ELTS
    float* scale1 = z + Z_ELTS;                 // 3520
    float* shift1 = scale1 + VC;                // 3520
    float* tmT    = shift1 + VC;                // 4096
    float* scale2 = tmT + 64 * 64;              // 64
    float* shift2 = scale2 + COUT;              // 64
    float* out    = (float*)d_out;

    shiftgcn_precompute<<<1, 256, 0, stream>>>(b, fm, g1, be1, mu1, vr1,
                                               g2, be2, mu2, vr2,
                                               scale1, shift1, tmT, scale2, shift2);

    shiftgcn_gemm_bn1<<<NT_TOT, 128, 0, stream>>>(x, W, tmT, scale1, shift1, z);

    shiftgcn_tail<<<NT_TOT, 256, 0, stream>>>(z, x, scale2, shift2, out);
}